// MSSSIM_33432025432686
// MI455X (gfx1250) — compile-verified
//
#include <hip/hip_runtime.h>
#include <math.h>

typedef __attribute__((ext_vector_type(2))) float v2f;
typedef __attribute__((ext_vector_type(8))) float v8f;

// 11-tap Gaussian, sigma=1.5, normalized (matches _gauss1d)
__constant__ float G11[11] = {
    0.00102838f, 0.00759874f, 0.03600077f, 0.10936113f, 0.21300553f,
    0.26601171f, 0.21300553f, 0.10936113f, 0.03600077f, 0.00759874f,
    0.00102838f};

__device__ __forceinline__ float bandv(int d) {
  return (d >= 0 && d <= 10) ? G11[d] : 0.0f;
}

// ---- dynamic LDS layout (floats) ----
#define SV1   0               // 32 x 28 raw v1 halo plane (pad rows/cols zero)
#define SV2   896             // 32 x 28 raw v2 halo plane
#define SXB   1792            // 5 x 28 x 16 x-blurred quantity planes
#define SRING 4032            // 5 x 11 x 256 ring of WH-blurred planes
#define SMEM_FLOATS 18112     // total
#define SMEM_BYTES  (SMEM_FLOATS * 4)

#define TPB 160               // 5 waves of 32: wave q handles quantity q

// Branch-free x-blur + y-blur for one quantity Q; qs is the SGPR-uniform wave id.
// x-blur: out(32x16) = A(data 32x28) * B(const band 28x16)
// y-blur: out(16x16) = A(const band 16x28) * B(xblur 28x16)
template <int Q>
__device__ __forceinline__ void msssim_phase_xy(
    float* __restrict__ sm, int qs, int zi, int m0, int kg, int hi8,
    const float coef0[7], const float coef1[7]) {
#pragma unroll
  for (int rb = 0; rb < 2; ++rb) {
    v8f acc = {0.f, 0.f, 0.f, 0.f, 0.f, 0.f, 0.f, 0.f};
    const int row = rb * 16 + m0;
#pragma unroll
    for (int i = 0; i < 7; ++i) {
      const int k0 = 4 * i + kg;
      float A0, A1;
      if (Q == 0) {
        A0 = sm[SV1 + row * 28 + k0];
        A1 = sm[SV1 + row * 28 + k0 + 1];
      } else if (Q == 1) {
        A0 = sm[SV2 + row * 28 + k0];
        A1 = sm[SV2 + row * 28 + k0 + 1];
      } else if (Q == 2) {
        float a0 = sm[SV1 + row * 28 + k0], a1 = sm[SV1 + row * 28 + k0 + 1];
        A0 = a0 * a0; A1 = a1 * a1;
      } else if (Q == 3) {
        float b0 = sm[SV2 + row * 28 + k0], b1 = sm[SV2 + row * 28 + k0 + 1];
        A0 = b0 * b0; A1 = b1 * b1;
      } else {
        float a0 = sm[SV1 + row * 28 + k0], a1 = sm[SV1 + row * 28 + k0 + 1];
        float b0 = sm[SV2 + row * 28 + k0], b1 = sm[SV2 + row * 28 + k0 + 1];
        A0 = a0 * b0; A1 = a1 * b1;
      }
      v2f Av = {A0, A1};
      v2f Bv = {coef0[i], coef1[i]};
      acc = __builtin_amdgcn_wmma_f32_16x16x4_f32(false, Av, false, Bv,
                                                  (short)0, acc, false, false);
    }
#pragma unroll
    for (int i = 0; i < 8; ++i) {
      int r = rb * 16 + i + hi8;
      if (r < 26) sm[SXB + (qs * 28 + r) * 16 + m0] = acc[i];
    }
  }

  {
    v8f acc = {0.f, 0.f, 0.f, 0.f, 0.f, 0.f, 0.f, 0.f};
#pragma unroll
    for (int i = 0; i < 7; ++i) {
      const int k0 = 4 * i + kg;
      v2f Av = {coef0[i], coef1[i]};
      v2f Bv = {sm[SXB + (qs * 28 + k0) * 16 + m0],
                sm[SXB + (qs * 28 + k0 + 1) * 16 + m0]};
      acc = __builtin_amdgcn_wmma_f32_16x16x4_f32(false, Av, false, Bv,
                                                  (short)0, acc, false, false);
    }
    const int slot = zi % 11;
#pragma unroll
    for (int i = 0; i < 8; ++i) {
      int r = i + hi8;
      sm[SRING + (qs * 11 + slot) * 256 + r * 16 + m0] = acc[i];
    }
  }
}

__global__ __launch_bounds__(TPB)
void msssim_ssim_kernel(const float* __restrict__ v1, const float* __restrict__ v2,
                        int D, int H, int W, int Dv, int Hv, int Wv, int ZS,
                        const int* __restrict__ flags, float* __restrict__ accum) {
  extern __shared__ float sm[];
  const int tid  = threadIdx.x;
  const int lane = tid & 31;
  // wave id == quantity id (0..4); force into an SGPR so dispatch is scalar
  const int qs   = __builtin_amdgcn_readfirstlane(tid >> 5);
  const int m0   = lane & 15;           // row (A) / col (B,C/D) within 16
  const int kg   = (lane >> 4) * 2;     // K sub-offset for lanes 16..31
  const int hi8  = (lane >> 4) * 8;     // C/D row offset for lanes 16..31

  // zero all LDS once (pad regions stay zero forever)
  for (int i = tid; i < SMEM_FLOATS; i += TPB) sm[i] = 0.0f;

  // per-lane banded-Toeplitz coefficients; the same matrix serves as
  // B (x-blur, band(k-col)) and A (y-blur, band(k-row)) since col==row==m0.
  float coef0[7], coef1[7];
#pragma unroll
  for (int i = 0; i < 7; ++i) {
    int k0 = 4 * i + kg;
    coef0[i] = bandv(k0 - m0);
    coef1[i] = bandv(k0 + 1 - m0);
  }
  float g[11];
#pragma unroll
  for (int t = 0; t < 11; ++t) g[t] = G11[t];

  const int fhi = flags[0];
  const int flo = flags[1];
  const float L  = (fhi ? 255.0f : 1.0f) + (flo ? 0.1f : 0.0f);
  const float C1 = (0.01f * L) * (0.01f * L);
  const float C2 = (0.03f * L) * (0.03f * L);

  const int b    = blockIdx.z / ZS;
  const int slab = blockIdx.z % ZS;
  const int zo0  = slab * 32;
  const int zo1  = min(zo0 + 32, Dv);
  const int x0   = blockIdx.x * 16;
  const int y0   = blockIdx.y * 16;

  float accS = 0.0f, accC = 0.0f;
  __syncthreads();  // LDS zeros visible

  for (int zi = zo0; zi < zo1 + 10; ++zi) {
    __syncthreads();  // nobody still reading staging/ring from prev plane

    // ---- Phase L: cooperative 26x26 halo load (zero-fill OOB) ----
    for (int e = tid; e < 26 * 26; e += TPB) {
      int r = e / 26, c = e % 26;
      int gy = y0 + r, gx = x0 + c;
      float a = 0.0f, bb = 0.0f;
      if (gy < H && gx < W) {
        size_t idx = ((size_t)(b * D + zi) * H + gy) * W + gx;
        a  = v1[idx];
        bb = v2[idx];
      }
      sm[SV1 + r * 28 + c] = a;
      sm[SV2 + r * 28 + c] = bb;
    }
    __syncthreads();

    // ---- Phases X+Y via WMMA, scalar dispatch on wave-uniform quantity id ----
    switch (qs) {
      case 0: msssim_phase_xy<0>(sm, qs, zi, m0, kg, hi8, coef0, coef1); break;
      case 1: msssim_phase_xy<1>(sm, qs, zi, m0, kg, hi8, coef0, coef1); break;
      case 2: msssim_phase_xy<2>(sm, qs, zi, m0, kg, hi8, coef0, coef1); break;
      case 3: msssim_phase_xy<3>(sm, qs, zi, m0, kg, hi8, coef0, coef1); break;
      default: msssim_phase_xy<4>(sm, qs, zi, m0, kg, hi8, coef0, coef1); break;
    }

    // ---- Phase Z: z-blur + SSIM for output plane zo = zi-10 ----
    if (zi >= zo0 + 10) {
      __syncthreads();  // all 5 rings for this plane visible to everyone
      int zo = zi - 10;
      int s0 = zo % 11;
      for (int p = tid; p < 256; p += TPB) {
        int ox = x0 + (p & 15);
        int oy = y0 + (p >> 4);
        if (ox < Wv && oy < Hv) {
          float zb[5];
#pragma unroll
          for (int qq = 0; qq < 5; ++qq) {
            float s = 0.0f;
#pragma unroll
            for (int t = 0; t < 11; ++t) {
              int slot = s0 + t; if (slot >= 11) slot -= 11;
              s += g[t] * sm[SRING + (qq * 11 + slot) * 256 + p];
            }
            zb[qq] = s;
          }
          float mu1 = zb[0], mu2 = zb[1];
          float mu1s = mu1 * mu1, mu2s = mu2 * mu2, mu12 = mu1 * mu2;
          float s1  = zb[2] - mu1s;
          float s2  = zb[3] - mu2s;
          float s12 = zb[4] - mu12;
          float va = 2.0f * s12 + C2;
          float vb = s1 + s2 + C2;
          float cs = va / vb;
          float ss = (2.0f * mu12 + C1) * va / ((mu1s + mu2s + C1) * vb);
          accS += ss;
          accC += cs;
        }
      }
    }
  }

  // wave reduce then one atomic per wave
#pragma unroll
  for (int off = 16; off > 0; off >>= 1) {
    accS += __shfl_down(accS, off);
    accC += __shfl_down(accC, off);
  }
  if (lane == 0) {
    atomicAdd(accum + 0, accS);
    atomicAdd(accum + 1, accC);
  }
}

// ---- data-dependent L flags: any(v1 > 128), any(v1 < -0.001) ----
__global__ void msssim_flags_kernel(const float* __restrict__ v, int n,
                                    int* __restrict__ flags) {
  __shared__ int sh[2];
  if (threadIdx.x == 0) { sh[0] = 0; sh[1] = 0; }
  __syncthreads();
  int hi = 0, lo = 0;
  for (int i = blockIdx.x * blockDim.x + threadIdx.x; i < n;
       i += gridDim.x * blockDim.x) {
    float x = v[i];
    hi |= (x > 128.0f);
    lo |= (x < -0.001f);
  }
  if (hi) atomicOr(&sh[0], 1);
  if (lo) atomicOr(&sh[1], 1);
  __syncthreads();
  if (threadIdx.x == 0) {
    if (sh[0]) atomicOr(&flags[0], 1);
    if (sh[1]) atomicOr(&flags[1], 1);
  }
}

// ---- 2x2x2 average pool for both volumes ----
__global__ void msssim_pool_kernel(const float* __restrict__ a,
                                   const float* __restrict__ b,
                                   float* __restrict__ oa, float* __restrict__ ob,
                                   int B, int D, int H, int W) {
  int Do = D >> 1, Ho = H >> 1, Wo = W >> 1;
  int n = B * Do * Ho * Wo;
  for (int i = blockIdx.x * blockDim.x + threadIdx.x; i < n;
       i += gridDim.x * blockDim.x) {
    int x = i % Wo; int t = i / Wo;
    int y = t % Ho; t /= Ho;
    int z = t % Do; int bb = t / Do;
    size_t p00 = ((size_t)(bb * D + 2 * z) * H + 2 * y) * W + 2 * x;
    size_t p01 = p00 + W;              // y+1
    size_t p10 = p00 + (size_t)H * W;  // z+1
    size_t p11 = p10 + W;
    float sa = a[p00] + a[p00 + 1] + a[p01] + a[p01 + 1] +
               a[p10] + a[p10 + 1] + a[p11] + a[p11 + 1];
    float sb = b[p00] + b[p00 + 1] + b[p01] + b[p01 + 1] +
               b[p10] + b[p10 + 1] + b[p11] + b[p11 + 1];
    oa[i] = sa * 0.125f;
    ob[i] = sb * 0.125f;
  }
}

__global__ void msssim_init_kernel(float* accum, int* flags) {
  int t = threadIdx.x;
  if (t < 8) { accum[t] = 0.0f; flags[t] = 0; }
}

// result = mcs0^w0 * mcs1^w1 * mcs2^w2 * (ssims3^w3)^3   (JAX broadcast semantics)
__global__ void msssim_final_kernel(const float* __restrict__ accum,
                                    float* __restrict__ out) {
  const float cnt[4] = {14281776.0f, 1503792.0f, 128304.0f, 5808.0f};
  const float wgt[4] = {0.0448f, 0.2856f, 0.3001f, 0.2363f};
  float ssims[4], mcs[4];
  for (int l = 0; l < 4; ++l) {
    ssims[l] = (accum[2 * l] / cnt[l] + 1.0f) * 0.5f;
    mcs[l]   = (accum[2 * l + 1] / cnt[l] + 1.0f) * 0.5f;
  }
  float p3 = powf(ssims[3], wgt[3]);
  float r = powf(mcs[0], wgt[0]) * powf(mcs[1], wgt[1]) * powf(mcs[2], wgt[2]);
  r *= p3 * p3 * p3;
  out[0] = r;
}

extern "C" void kernel_launch(void* const* d_in, const int* in_sizes, int n_in,
                              void* d_out, int out_size, void* d_ws, size_t ws_size,
                              hipStream_t stream) {
  (void)in_sizes; (void)n_in; (void)out_size; (void)ws_size;
  const float* v1 = (const float*)d_in[0];
  const float* v2 = (const float*)d_in[1];
  float* out = (float*)d_out;
  char* wsb = (char*)d_ws;
  float* accum = (float*)wsb;              // 8 floats: {ssim,cs} per level
  int*   flags = (int*)(wsb + 64);         // 2 ints per level
  float* pool  = (float*)(wsb + 256);
  const int B = 2;
  size_t s1 = (size_t)B * 64 * 128 * 128;
  size_t s2 = (size_t)B * 32 * 64 * 64;
  size_t s3 = (size_t)B * 16 * 32 * 32;
  float* a1 = pool;      float* b1 = a1 + s1;
  float* a2 = b1 + s1;   float* b2 = a2 + s2;
  float* a3 = b2 + s2;   float* b3 = a3 + s3;

  msssim_init_kernel<<<1, 32, 0, stream>>>(accum, flags);

  const float* LA[4] = {v1, a1, a2, a3};
  const float* LB[4] = {v2, b1, b2, b3};
  float* PA[3] = {a1, a2, a3};
  float* PB[3] = {b1, b2, b3};

  for (int l = 0; l < 4; ++l) {
    int D = 128 >> l, H = 256 >> l, W = 256 >> l;
    int n = B * D * H * W;
    int Dv = D - 10, Hv = H - 10, Wv = W - 10;
    int TX = (Wv + 15) / 16, TY = (Hv + 15) / 16, ZS = (Dv + 31) / 32;

    int fb = (n + 255) / 256; if (fb > 2048) fb = 2048;
    msssim_flags_kernel<<<fb, 256, 0, stream>>>(LA[l], n, flags + 2 * l);

    dim3 grid((unsigned)TX, (unsigned)TY, (unsigned)(B * ZS));
    msssim_ssim_kernel<<<grid, TPB, SMEM_BYTES, stream>>>(
        LA[l], LB[l], D, H, W, Dv, Hv, Wv, ZS, flags + 2 * l, accum + 2 * l);

    if (l < 3) {
      int no = B * (D / 2) * (H / 2) * (W / 2);
      int pb = (no + 255) / 256; if (pb > 4096) pb = 4096;
      msssim_pool_kernel<<<pb, 256, 0, stream>>>(LA[l], LB[l], PA[l], PB[l],
                                                 B, D, H, W);
    }
  }
  msssim_final_kernel<<<1, 1, 0, stream>>>(accum, out);
}